// SGConv_17489106829753
// MI455X (gfx1250) — compile-verified
//
#include <hip/hip_runtime.h>
#include <hip/hip_bf16.h>

// ---------------------------------------------------------------------------
// SGConv: z = relu(concat_c( A_c @ (x @ W[:,c,:]) ))
//   x   : [N, 256]  f32
//   W   : [256, 4, 32] f32  (flattened row-major == [256, 128])
//   rows/cols : [4, E] int32, vals : [4, E] f32
//   out : [N, 128] f32
// Stage 1: tmp = x @ W  via V_WMMA_F32_16X16X4_F32  -> d_ws ([N,128] f32)
// Stage 2: edge-parallel gather/scatter-add with GLOBAL_ATOMIC_ADD_F32
// Stage 3: in-place ReLU
// ---------------------------------------------------------------------------

typedef __attribute__((ext_vector_type(2))) float v2f;
typedef __attribute__((ext_vector_type(8))) float v8f;

#define D_IN   256
#define D_OUT  128   // C * D_C
#define D_C    32

// ---- Stage 1: WMMA f32 GEMM --------------------------------------------
// Block = 256 threads = 8 waves. Block computes a 16-row x 128-col tile.
// Wave w computes the 16x16 tile at columns [16w, 16w+16).
// A (16x4 f32) layout: lanes 0-15 -> M=lane, K = k0+{0,1} in v[0],v[1]
//                      lanes 16-31 -> M=lane-16, K = k0+{2,3}
// B (4x16 f32) layout: lanes 0-15 -> N=lane, K = k0+{0,1}
//                      lanes 16-31 -> N=lane-16, K = k0+{2,3}
// C/D (16x16 f32): VGPR r: lanes 0-15 -> M=r, N=lane; lanes 16-31 -> M=r+8.
__global__ __launch_bounds__(256)
void sgconv_gemm_wmma(const float* __restrict__ x,
                      const float* __restrict__ W,
                      float* __restrict__ tmp,
                      int nrows) {
  const int lane    = threadIdx.x & 31;
  const int wave    = threadIdx.x >> 5;          // 0..7 -> column tile
  const int halfSel = lane >> 4;                 // 0 or 1
  const int l       = lane & 15;

  const int rowBase = blockIdx.x * 16;
  const int colBase = wave * 16;
  if (rowBase >= nrows) return;

  const float* __restrict__ xrow = x + (size_t)(rowBase + l) * D_IN + 2 * halfSel;
  const float* __restrict__ wcol = W + (size_t)(2 * halfSel) * D_OUT + colBase + l;

  v8f acc = {};
#pragma unroll 8
  for (int k = 0; k < D_IN; k += 4) {
    // A fragment: two consecutive K elements of this lane's row (8B aligned)
    v2f a = *(const v2f*)(xrow + k);
    // B fragment: two K-rows of W at this lane's column (stride D_OUT)
    v2f b;
    b.x = wcol[(size_t)k * D_OUT];
    b.y = wcol[(size_t)k * D_OUT + D_OUT];
    acc = __builtin_amdgcn_wmma_f32_16x16x4_f32(
        /*neg_a=*/false, a, /*neg_b=*/false, b,
        /*c_mod=*/(short)0, acc, /*reuse_a=*/false, /*reuse_b=*/false);
  }

  float* __restrict__ outp =
      tmp + (size_t)(rowBase + halfSel * 8) * D_OUT + colBase + l;
#pragma unroll
  for (int r = 0; r < 8; ++r) {
    outp[(size_t)r * D_OUT] = acc[r];
  }
}

// ---- Stage 2: edge-parallel SpMM scatter-add ---------------------------
// lane = feature (32 features per class); one edge per wave; grid-stride.
__global__ __launch_bounds__(256)
void sgconv_spmm(const float* __restrict__ tmp,
                 const int*   __restrict__ rows,
                 const int*   __restrict__ cols,
                 const float* __restrict__ vals,
                 float*       __restrict__ out,
                 long long totalEdges, int E) {
  const int lane = threadIdx.x & 31;
  long long waveId =
      (((long long)blockIdx.x * blockDim.x) + threadIdx.x) >> 5;
  const long long nWaves = ((long long)gridDim.x * blockDim.x) >> 5;

  for (long long e = waveId; e < totalEdges; e += nWaves) {
    const int c  = (int)(e / E);                 // class id
    const int r  = rows[e];
    const int cl = cols[e];
    const float v = vals[e];
    const float m = tmp[(size_t)cl * D_OUT + c * D_C + lane] * v;
    atomicAdd(out + (size_t)r * D_OUT + c * D_C + lane, m);
  }
}

// ---- Stage 3: in-place ReLU --------------------------------------------
__global__ __launch_bounds__(256)
void sgconv_relu(float* __restrict__ out, long long n) {
  long long i = (long long)blockIdx.x * blockDim.x + threadIdx.x;
  if (i < n) {
    float v = out[i];
    out[i] = v > 0.0f ? v : 0.0f;
  }
}

extern "C" void kernel_launch(void* const* d_in, const int* in_sizes, int n_in,
                              void* d_out, int out_size, void* d_ws, size_t ws_size,
                              hipStream_t stream) {
  const float* x    = (const float*)d_in[0];   // [N, 256]
  const float* W    = (const float*)d_in[1];   // [256, 128]
  const int*   rows = (const int*)d_in[2];     // [4, E]
  const int*   cols = (const int*)d_in[3];     // [4, E]
  const float* vals = (const float*)d_in[4];   // [4, E]
  float* out = (float*)d_out;                  // [N, 128]
  float* tmp = (float*)d_ws;                   // [N, 128] scratch

  const int N = in_sizes[0] / D_IN;
  const long long totalEdges = in_sizes[2];    // C * E
  const int E = (int)(totalEdges / 4);

  // Stage 1: tmp = x @ W (WMMA f32)
  {
    dim3 grid((N + 15) / 16);
    dim3 block(256);
    sgconv_gemm_wmma<<<grid, block, 0, stream>>>(x, W, tmp, N);
  }

  // Zero the accumulator (d_out) before atomic scatter.
  hipMemsetAsync(d_out, 0, (size_t)out_size * sizeof(float), stream);

  // Stage 2: scatter-add. One wave per edge, grid-stride.
  {
    const int blocks = 16384;                  // 131072 waves; ~49 edges/wave
    sgconv_spmm<<<blocks, 256, 0, stream>>>(tmp, rows, cols, vals, out,
                                            totalEdges, E);
  }

  // Stage 3: ReLU in place.
  {
    long long n = (long long)out_size;
    int blocks = (int)((n + 255) / 256);
    sgconv_relu<<<blocks, 256, 0, stream>>>(out, n);
  }
}